// DirectionalGINConv_19610820673953
// MI455X (gfx1250) — compile-verified
//
#include <hip/hip_runtime.h>

typedef __attribute__((ext_vector_type(2))) float v2f;
typedef __attribute__((ext_vector_type(8))) float v8f;

#define N_NODES 50000
#define N_EDGES 800000
#define CH 64
#define NTILES (N_NODES / 16)   // 3125, exact
#define LDSW_STRIDE 65          // pad to avoid 16-way LDS bank conflicts on stride-64 rows

// h = x  (GIN eps=0: h_i starts at x_i, scatter adds neighbors on top)
__global__ void init_h_kernel(const float4* __restrict__ x, float4* __restrict__ h, int n4) {
    int i = blockIdx.x * blockDim.x + threadIdx.x;
    if (i < n4) h[i] = x[i];
}

// h[dst, c] += x[src, c] for every edge; one thread per (edge, channel).
// unsafeAtomicAdd -> native GLOBAL_ATOMIC_ADD_F32 (no CAS loop); RMW resolved
// at the L2 atomic units on the L2-resident 12.8 MB aggregate buffer.
__global__ void scatter_add_kernel(const float* __restrict__ x,
                                   const long long* __restrict__ edge_index,
                                   float* __restrict__ h) {
    long long i = (long long)blockIdx.x * blockDim.x + threadIdx.x;
    int e = (int)(i >> 6);
    int c = (int)(i & 63);
    if (e < N_EDGES) {
        int s = (int)edge_index[e];            // row 0: src
        int d = (int)edge_index[N_EDGES + e];  // row 1: dst
        unsafeAtomicAdd(&h[(size_t)d * CH + c], x[(size_t)s * CH + c]);
    }
}

// out[n, o] = relu( sum_i h[n,i] * W[o,i] + b[o] )   via V_WMMA_F32_16X16X4_F32
// One wave per 16-node tile; 4 v8f accumulators cover the 64 output channels.
__global__ void gin_mlp_wmma_kernel(const float* __restrict__ h,
                                    const float* __restrict__ W,
                                    const float* __restrict__ b,
                                    float* __restrict__ out) {
    __shared__ float lW[CH * LDSW_STRIDE];

    // Block-cooperative stage of W (64x64 = 16KB) into LDS, padded stride.
    for (int i = threadIdx.x; i < CH * CH; i += blockDim.x) {
        int r = i >> 6;
        int c = i & 63;
        lW[r * LDSW_STRIDE + c] = W[i];
    }
    __syncthreads();

    int lane = threadIdx.x & 31;
    int wave = threadIdx.x >> 5;
    int tile = blockIdx.x * (blockDim.x >> 5) + wave;
    if (tile >= NTILES) return;   // whole-wave uniform: EXEC stays all-1s for WMMA

    int half = lane >> 4;     // 0: K pair {k0,k0+1}; 1: {k0+2,k0+3}
    int l16  = lane & 15;
    int node0 = tile * 16;

    // A-matrix (16x4 f32) per ISA layout: lane = half*16 + m, VGPR j = A[m, 2*half + j]
    const float* hrow = h + (size_t)(node0 + l16) * CH + 2 * half;

    v8f acc[4] = {};   // 4 N-tiles of 16 output channels each

#pragma unroll
    for (int k0 = 0; k0 < CH; k0 += 4) {
        v2f a;
        a.x = hrow[k0];
        a.y = hrow[k0 + 1];
#pragma unroll
        for (int t = 0; t < 4; ++t) {
            // B-matrix (4x16 f32): lane = half*16 + n, VGPR j = B[2*half + j, n] = W[o_base+n, k0+2*half+j]
            const float* wp = &lW[(t * 16 + l16) * LDSW_STRIDE + k0 + 2 * half];
            v2f bm;
            bm.x = wp[0];
            bm.y = wp[1];
            acc[t] = __builtin_amdgcn_wmma_f32_16x16x4_f32(
                /*neg_a=*/false, a, /*neg_b=*/false, bm,
                /*c_mod=*/(short)0, acc[t], /*reuse_a=*/false, /*reuse_b=*/false);
        }
    }

    // Epilogue: bias + ReLU (relu(relu(x)) == relu(x)), store per C/D layout:
    // VGPR v holds row M = v + 8*half, column N = lane%16.
#pragma unroll
    for (int t = 0; t < 4; ++t) {
        float bias = b[t * 16 + l16];
#pragma unroll
        for (int v = 0; v < 8; ++v) {
            int row = node0 + v + 8 * half;
            float val = acc[t][v] + bias;
            out[(size_t)row * CH + t * 16 + l16] = fmaxf(val, 0.f);
        }
    }
}

extern "C" void kernel_launch(void* const* d_in, const int* in_sizes, int n_in,
                              void* d_out, int out_size, void* d_ws, size_t ws_size,
                              hipStream_t stream) {
    const float*     x  = (const float*)d_in[0];
    const long long* ei = (const long long*)d_in[1];   // int64 edge_index (2, E)
    const float*     W  = (const float*)d_in[2];
    const float*     b  = (const float*)d_in[3];
    float* out = (float*)d_out;
    float* h   = (float*)d_ws;   // 50000*64 f32 = 12.8 MB scratch

    // 1) h = x
    int n4 = N_NODES * CH / 4;   // 800000 float4s
    init_h_kernel<<<(n4 + 255) / 256, 256, 0, stream>>>((const float4*)x, (float4*)h, n4);

    // 2) h[dst] += x[src]  (51.2M native f32 atomics, coalesced per edge)
    long long total = (long long)N_EDGES * CH;           // 51,200,000
    int blocks2 = (int)((total + 255) / 256);            // 200,000
    scatter_add_kernel<<<blocks2, 256, 0, stream>>>(x, ei, h);

    // 3) WMMA GEMM + bias + ReLU; 8 waves/block, one 16-node tile per wave
    int blocks3 = (NTILES + 7) / 8;                      // 391
    gin_mlp_wmma_kernel<<<blocks3, 256, 0, stream>>>(h, W, b, out);
}